// VectorQuantizer_87265145520455
// MI455X (gfx1250) — compile-verified
//
#include <hip/hip_runtime.h>

typedef __attribute__((ext_vector_type(2))) float v2f;
typedef __attribute__((ext_vector_type(8))) float v8f;

#define C_DIM   128
#define K_DIM   512
#define N_TOTAL (32 * 128 * 64 * 64)   // 16,777,216 elements of z_e / z_q
#define N4      (N_TOTAL / 4)
#define NB2     4096                   // blocks for the streaming pass
#define TPB     256                    // 8 waves (wave32)

// ---------------------------------------------------------------------------
// Phase 1: row norms of embedding via V_WMMA_F32_16X16X4_F32 (diag of E*E^T),
// then a block-wide argmin. One block, 8 waves; each wave owns 4 tiles of
// 16 rows (8 * 4 * 16 = 512 rows). For each 16x4 chunk the same v2f register
// pair is fed as both A and B (A-layout: lane<16 holds K=0,1; lane>=16 holds
// K=2,3 for rows M=lane&15 -> B interpretation is exactly A^T), so the D
// diagonal accumulates sum_c e[r][c]^2 across the 32 chunks of C=128.
// Diagonal lives at (VGPR i, lane i) for i<8 and (VGPR i-8, lane 16+i) for
// i>=8 per the 16x16 f32 C/D layout.
// ---------------------------------------------------------------------------
__global__ void vq_argmin_kernel(const float* __restrict__ emb,
                                 int* __restrict__ kstar)
{
    __shared__ float s_norm[K_DIM];
    __shared__ float s_val[TPB];
    __shared__ int   s_idx[TPB];

    const int tid  = threadIdx.x;
    const int wave = tid >> 5;
    const int lane = tid & 31;

    const int rowInTile = lane & 15;
    const int colPair   = (lane >> 4) << 1;   // 0 for lanes 0-15, 2 for 16-31

    for (int t = 0; t < 4; ++t) {
        const int r0  = wave * 64 + t * 16;
        const int row = r0 + rowInTile;
        v8f acc = {};
        for (int c0 = 0; c0 < C_DIM; c0 += 4) {
            v2f a = *(const v2f*)(emb + row * C_DIM + c0 + colPair);
            acc = __builtin_amdgcn_wmma_f32_16x16x4_f32(
                false, a, false, a, (short)0, acc, false, false);
        }
        if (lane < 8) {
            s_norm[r0 + lane] = acc[lane];            // diag i = lane (0..7)
        } else if (lane >= 24) {
            s_norm[r0 + (lane - 16)] = acc[lane - 24]; // diag i = lane-16 (8..15)
        }
    }
    __syncthreads();

    // argmin over 512 norms; strict '<' keeps the lowest index on ties
    // (matches jnp.argmin first-occurrence semantics).
    float v  = s_norm[tid];
    int   bi = tid;
    float v2 = s_norm[tid + TPB];
    if (v2 < v) { v = v2; bi = tid + TPB; }
    s_val[tid] = v;
    s_idx[tid] = bi;
    __syncthreads();
    for (int s = TPB / 2; s > 0; s >>= 1) {
        if (tid < s) {
            if (s_val[tid + s] < s_val[tid]) {
                s_val[tid] = s_val[tid + s];
                s_idx[tid] = s_idx[tid + s];
            }
        }
        __syncthreads();
    }
    if (tid == 0) *kstar = s_idx[0];
}

// ---------------------------------------------------------------------------
// Phase 2: streaming pass. z_q[b,c,h,w] = emb[k*, c]; accumulate (e - z)^2.
// float4 loads/stores (HW plane = 4096 floats, so a float4 never crosses a
// c-boundary); selected code row broadcast through LDS; prefetch one
// grid-stride ahead (global_prefetch_b8). Fixed grid -> deterministic
// per-block partial sums (no float atomics).
// ---------------------------------------------------------------------------
__global__ void vq_quantize_kernel(const float* __restrict__ z_e,
                                   const float* __restrict__ emb,
                                   const int* __restrict__ kstar,
                                   float* __restrict__ z_q,
                                   float* __restrict__ partials)
{
    __shared__ float s_row[C_DIM];
    __shared__ float s_red[TPB];

    const int tid = threadIdx.x;
    const int k   = *kstar;
    if (tid < C_DIM) s_row[tid] = emb[k * C_DIM + tid];
    __syncthreads();

    const float4* zin  = (const float4*)z_e;
    float4*       zout = (float4*)z_q;
    const int stride = gridDim.x * blockDim.x;

    float acc = 0.f;
    for (int i = blockIdx.x * blockDim.x + tid; i < N4; i += stride) {
        __builtin_prefetch((const void*)(zin + i + stride), 0, 1);
        float4 z = zin[i];
        // float4 index -> plane index = i >> 10 (1024 float4 per HW plane);
        // channel = plane % 128.
        float e = s_row[(i >> 10) & (C_DIM - 1)];
        float4 q; q.x = e; q.y = e; q.z = e; q.w = e;
        zout[i] = q;
        float d0 = e - z.x, d1 = e - z.y, d2 = e - z.z, d3 = e - z.w;
        acc += d0 * d0 + d1 * d1 + d2 * d2 + d3 * d3;
    }

    s_red[tid] = acc;
    __syncthreads();
    for (int s = TPB / 2; s > 0; s >>= 1) {
        if (tid < s) s_red[tid] += s_red[tid + s];
        __syncthreads();
    }
    if (tid == 0) partials[blockIdx.x] = s_red[0];
}

// ---------------------------------------------------------------------------
// Phase 3: deterministic final reduce of the NB2 partials;
// vq_loss = codebook + commitment = 2 * sum / N.
// ---------------------------------------------------------------------------
__global__ void vq_loss_kernel(const float* __restrict__ partials,
                               float* __restrict__ loss_out)
{
    __shared__ float s_red[TPB];
    const int tid = threadIdx.x;
    float acc = 0.f;
    for (int i = tid; i < NB2; i += TPB) acc += partials[i];
    s_red[tid] = acc;
    __syncthreads();
    for (int s = TPB / 2; s > 0; s >>= 1) {
        if (tid < s) s_red[tid] += s_red[tid + s];
        __syncthreads();
    }
    if (tid == 0) *loss_out = 2.0f * s_red[0] / (float)N_TOTAL;
}

extern "C" void kernel_launch(void* const* d_in, const int* in_sizes, int n_in,
                              void* d_out, int out_size, void* d_ws, size_t ws_size,
                              hipStream_t stream)
{
    const float* z_e = (const float*)d_in[0];   // [32,128,64,64] f32
    const float* emb = (const float*)d_in[1];   // [512,128] f32

    float* out      = (float*)d_out;            // [N_TOTAL z_q][1 vq_loss]
    int*   kstar    = (int*)d_ws;               // ws[0..3]: selected index
    float* partials = (float*)d_ws + 16;        // 64B-offset: NB2 partial sums

    hipLaunchKernelGGL(vq_argmin_kernel, dim3(1), dim3(TPB), 0, stream,
                       emb, kstar);
    hipLaunchKernelGGL(vq_quantize_kernel, dim3(NB2), dim3(TPB), 0, stream,
                       z_e, emb, kstar, out, partials);
    hipLaunchKernelGGL(vq_loss_kernel, dim3(1), dim3(TPB), 0, stream,
                       partials, out + N_TOTAL);
}